// UniADetHead_24945170055318
// MI455X (gfx1250) — compile-verified
//
#include <hip/hip_runtime.h>
#include <math.h>

#define TAU   0.07f
#define EPSN  1e-12f

typedef __attribute__((ext_vector_type(2))) float v2f;
typedef __attribute__((ext_vector_type(4))) float v4f;
typedef __attribute__((ext_vector_type(8))) float v8f;

__device__ __forceinline__ int imin(int a, int b) { return a < b ? a : b; }

// ---------------------------------------------------------------------------
// Prep: ||w_seg[l][c]|| for 4 layers x 2 classes -> ws[0..7]
// ---------------------------------------------------------------------------
__global__ __launch_bounds__(256)
void wnorm_kernel(const float* __restrict__ w0, const float* __restrict__ w1,
                  const float* __restrict__ w2, const float* __restrict__ w3,
                  float* __restrict__ wsn) {
    __shared__ float red[256];
    const int layer = blockIdx.x >> 1;
    const int cls   = blockIdx.x & 1;
    const float* w = (layer == 0 ? w0 : layer == 1 ? w1 : layer == 2 ? w2 : w3) + cls * 1024;
    float s = 0.f;
    for (int i = threadIdx.x; i < 1024; i += 256) { float v = w[i]; s += v * v; }
    red[threadIdx.x] = s;
    __syncthreads();
    for (int off = 128; off > 0; off >>= 1) {
        if (threadIdx.x < off) red[threadIdx.x] += red[threadIdx.x + off];
        __syncthreads();
    }
    if (threadIdx.x == 0) wsn[blockIdx.x] = fmaxf(sqrtf(red[0]), EPSN);
}

// ---------------------------------------------------------------------------
// cls logits: [4,32,2] -> d_out[0..255]
// logit_c = dot(x, w_c) / (max(||x||,eps) * max(||w_c||,eps) * tau)
// ---------------------------------------------------------------------------
__global__ __launch_bounds__(256)
void cls_kernel(const float* __restrict__ c0, const float* __restrict__ c1,
                const float* __restrict__ c2, const float* __restrict__ c3,
                const float* __restrict__ w0, const float* __restrict__ w1,
                const float* __restrict__ w2, const float* __restrict__ w3,
                float* __restrict__ out) {
    __shared__ float red[5][256];
    const int img = blockIdx.x;            // l*32 + b
    const int layer = img >> 5, b = img & 31;
    const float* x = (layer == 0 ? c0 : layer == 1 ? c1 : layer == 2 ? c2 : c3) + b * 1024;
    const float* w = (layer == 0 ? w0 : layer == 1 ? w1 : layer == 2 ? w2 : w3);
    float sx = 0.f, d0 = 0.f, d1 = 0.f, s0 = 0.f, s1 = 0.f;
    for (int i = threadIdx.x; i < 1024; i += 256) {
        float xv = x[i], a = w[i], bb = w[1024 + i];
        sx += xv * xv; d0 += xv * a; d1 += xv * bb; s0 += a * a; s1 += bb * bb;
    }
    const int tid = threadIdx.x;
    red[0][tid] = sx; red[1][tid] = d0; red[2][tid] = d1; red[3][tid] = s0; red[4][tid] = s1;
    __syncthreads();
    for (int off = 128; off > 0; off >>= 1) {
        if (tid < off) {
            red[0][tid] += red[0][tid + off];
            red[1][tid] += red[1][tid + off];
            red[2][tid] += red[2][tid + off];
            red[3][tid] += red[3][tid + off];
            red[4][tid] += red[4][tid + off];
        }
        __syncthreads();
    }
    if (tid == 0) {
        float nx = fmaxf(sqrtf(red[0][0]), EPSN);
        float n0 = fmaxf(sqrtf(red[3][0]), EPSN);
        float n1 = fmaxf(sqrtf(red[4][0]), EPSN);
        out[img * 2 + 0] = red[1][0] / (nx * n0 * TAU);
        out[img * 2 + 1] = red[2][0] / (nx * n1 * TAU);
    }
}

// ---------------------------------------------------------------------------
// Stage 1: per-row cosine logits via V_WMMA_F32_16X16X4_F32, fused softmax.
// Each wave owns a 16-row tile; K marches in steps of 8 (two WMMAs).
// A frag (32-bit A 16x4) loaded DIRECTLY: lane L (r=L%16, half=L/16) loads
// float2 at (row r, col k + 2*half) -> exactly (VGPR0, VGPR1) of the frag.
// No cross-lane shuffles needed. Sum-of-squares: lanes r and r+16 jointly
// cover all K; one xor-16 reduce at the end.
// B frag: columns 0/1 = w_seg class vectors from a 3-row LDS tile whose
// third row is zeros (so no mask multiplies, no divergent loads; EXEC
// stays all-ones for WMMA).
// probs[layer*32768 + row] = softmax class-0 prob of the two seg logits.
// ---------------------------------------------------------------------------
__global__ __launch_bounds__(256)
void seg_wmma_kernel(const float* __restrict__ p0, const float* __restrict__ p1,
                     const float* __restrict__ p2, const float* __restrict__ p3,
                     const float* __restrict__ w0, const float* __restrict__ w1,
                     const float* __restrict__ w2, const float* __restrict__ w3,
                     const float* __restrict__ wsn, float* __restrict__ probs) {
    __shared__ float ldsW[3072];            // w_seg [2][1024] + zero row
    __shared__ float ldsD[8][16][2];        // per-wave dot extraction

    const int layer = blockIdx.y;
    const float* patch = (layer == 0 ? p0 : layer == 1 ? p1 : layer == 2 ? p2 : p3);
    const float* w     = (layer == 0 ? w0 : layer == 1 ? w1 : layer == 2 ? w2 : w3);

    for (int i = threadIdx.x; i < 3072; i += 256) ldsW[i] = (i < 2048) ? w[i] : 0.f;
    __syncthreads();

    const float wn0 = wsn[layer * 2 + 0];
    const float wn1 = wsn[layer * 2 + 1];

    const int wave = threadIdx.x >> 5;
    const int lane = threadIdx.x & 31;
    const int r    = lane & 15;             // A: row M; B: column N
    const int half = lane >> 4;
    const int cb   = 2 * half;              // K sub-offset for this half
    const int tileRow0 = (blockIdx.x * 8 + wave) * 16;

    const float* row = patch + (size_t)(tileRow0 + r) * 1024;
    const float* wc  = &ldsW[imin(r, 2) * 1024];   // rows >=2 read zeros

    v8f acc = {};
    float ss = 0.f;

    for (int k = 0; k < 1024; k += 8) {
        const v2f qa = __builtin_nontemporal_load(
            reinterpret_cast<const v2f*>(row + k + cb));
        const v2f qb = __builtin_nontemporal_load(
            reinterpret_cast<const v2f*>(row + k + 4 + cb));
        __builtin_prefetch(row + k + 256, 0, 1);
        ss += qa[0] * qa[0] + qa[1] * qa[1] + qb[0] * qb[0] + qb[1] * qb[1];

        v2f b1; b1[0] = wc[k + cb];     b1[1] = wc[k + cb + 1];
        acc = __builtin_amdgcn_wmma_f32_16x16x4_f32(false, qa, false, b1,
                                                    (short)0, acc, false, false);
        v2f b2; b2[0] = wc[k + 4 + cb]; b2[1] = wc[k + 5 + cb];
        acc = __builtin_amdgcn_wmma_f32_16x16x4_f32(false, qb, false, b2,
                                                    (short)0, acc, false, false);
    }

    ss += __shfl_xor(ss, 16, 32);           // full ||row||^2 on every lane

    // D layout: lane L, VGPR v -> D[M = v + 8*(L/16)][N = L%16]
    if (r < 2) {
        const int mb = half * 8;
        ldsD[wave][mb + 0][r] = acc[0];
        ldsD[wave][mb + 1][r] = acc[1];
        ldsD[wave][mb + 2][r] = acc[2];
        ldsD[wave][mb + 3][r] = acc[3];
        ldsD[wave][mb + 4][r] = acc[4];
        ldsD[wave][mb + 5][r] = acc[5];
        ldsD[wave][mb + 6][r] = acc[6];
        ldsD[wave][mb + 7][r] = acc[7];
    }
    __syncthreads();

    if (lane < 16) {
        const float dot0 = ldsD[wave][lane][0];
        const float dot1 = ldsD[wave][lane][1];
        const float denom = fmaxf(sqrtf(ss), EPSN);
        const float l0 = dot0 / (denom * wn0 * TAU);
        const float l1 = dot1 / (denom * wn1 * TAU);
        const float pr = 1.f / (1.f + expf(l1 - l0));   // softmax class-0 prob
        probs[(size_t)layer * 32768 + tileRow0 + lane] = pr;
    }
}

// ---------------------------------------------------------------------------
// Stage 2: bilinear 32x32 -> 512x512 (half-pixel), 2 channels (p, 1-p).
// Map kept in LDS; non-temporal float4 stores (268 MB streamed once, > L2).
// ---------------------------------------------------------------------------
__global__ __launch_bounds__(256)
void upsample_kernel(const float* __restrict__ probs, float* __restrict__ out) {
    __shared__ float m[1024];
    const int img = blockIdx.y;             // l*32 + b, 0..127
    const float* src = probs + (size_t)img * 1024;
    for (int i = threadIdx.x; i < 1024; i += 256) m[i] = src[i];
    __syncthreads();

    float* out0 = out + (size_t)img * 2 * 262144;
    float* out1 = out0 + 262144;

    for (int it = 0; it < 16; ++it) {
        const int g  = blockIdx.x * 4096 + it * 256 + threadIdx.x; // 4-px group
        const int y  = g >> 7;
        const int xb = (g & 127) << 2;

        float sy = fmaxf((y + 0.5f) * 0.0625f - 0.5f, 0.f);
        int   y0 = (int)sy;
        float fy = sy - (float)y0;
        int   y1 = imin(y0 + 1, 31);
        y0 = imin(y0, 31);
        const float* r0 = m + y0 * 32;
        const float* r1 = m + y1 * 32;

        float vv[4];
#pragma unroll
        for (int j = 0; j < 4; ++j) {
            float sx = fmaxf((xb + j + 0.5f) * 0.0625f - 0.5f, 0.f);
            int   x0 = (int)sx;
            float fx = sx - (float)x0;
            int   x1 = imin(x0 + 1, 31);
            x0 = imin(x0, 31);
            const float top = r0[x0] + fx * (r0[x1] - r0[x0]);
            const float bot = r1[x0] + fx * (r1[x1] - r1[x0]);
            vv[j] = top + fy * (bot - top);
        }
        v4f o0; o0[0] = vv[0]; o0[1] = vv[1]; o0[2] = vv[2]; o0[3] = vv[3];
        v4f o1; o1[0] = 1.f - vv[0]; o1[1] = 1.f - vv[1];
        o1[2] = 1.f - vv[2]; o1[3] = 1.f - vv[3];
        __builtin_nontemporal_store(o0, reinterpret_cast<v4f*>(out0 + y * 512 + xb));
        __builtin_nontemporal_store(o1, reinterpret_cast<v4f*>(out1 + y * 512 + xb));
    }
}

// ---------------------------------------------------------------------------
extern "C" void kernel_launch(void* const* d_in, const int* in_sizes, int n_in,
                              void* d_out, int out_size, void* d_ws, size_t ws_size,
                              hipStream_t stream) {
    (void)in_sizes; (void)n_in; (void)out_size; (void)ws_size;
    // setup_inputs order: per layer l in {5,11,17,23}: cls, patch, w_cls, w_seg
    const float* cls5   = (const float*)d_in[0];
    const float* patch5 = (const float*)d_in[1];
    const float* wcls5  = (const float*)d_in[2];
    const float* wseg5  = (const float*)d_in[3];
    const float* cls11   = (const float*)d_in[4];
    const float* patch11 = (const float*)d_in[5];
    const float* wcls11  = (const float*)d_in[6];
    const float* wseg11  = (const float*)d_in[7];
    const float* cls17   = (const float*)d_in[8];
    const float* patch17 = (const float*)d_in[9];
    const float* wcls17  = (const float*)d_in[10];
    const float* wseg17  = (const float*)d_in[11];
    const float* cls23   = (const float*)d_in[12];
    const float* patch23 = (const float*)d_in[13];
    const float* wcls23  = (const float*)d_in[14];
    const float* wseg23  = (const float*)d_in[15];

    float* out   = (float*)d_out;
    float* wsf   = (float*)d_ws;
    float* wsn   = wsf;          // 8 floats: seg weight norms
    float* probs = wsf + 16;     // 4*32768 floats: class-0 prob maps

    wnorm_kernel<<<dim3(8), 256, 0, stream>>>(wseg5, wseg11, wseg17, wseg23, wsn);
    cls_kernel<<<dim3(128), 256, 0, stream>>>(cls5, cls11, cls17, cls23,
                                              wcls5, wcls11, wcls17, wcls23, out);
    seg_wmma_kernel<<<dim3(256, 4), 256, 0, stream>>>(patch5, patch11, patch17, patch23,
                                                      wseg5, wseg11, wseg17, wseg23,
                                                      wsn, probs);
    upsample_kernel<<<dim3(16, 128), 256, 0, stream>>>(probs, out + 256);
}